// PhaseSyncAttention_48722109005882
// MI455X (gfx1250) — compile-verified
//
#include <hip/hip_runtime.h>
#include <math.h>

typedef __attribute__((ext_vector_type(2))) float v2f;
typedef __attribute__((ext_vector_type(8))) float v8f;

#define B_DIM 8
#define L_DIM 1024
#define H_DIM 8
#define GAMMA 2.0f

// ---------------------------------------------------------------------------
// Kernel 1: psi[bh][l] = cumsum_l exp(GAMMA * clip(hurst[b,l,h], 0.01, 1))
// One block of 256 threads per (b,h); each thread owns 4 contiguous elements;
// Hillis-Steele scan over the 256 chunk sums in LDS.
// ---------------------------------------------------------------------------
__global__ __launch_bounds__(256) void psi_cumsum_kernel(
    const float* __restrict__ hurst, float* __restrict__ psi) {
  const int bh = blockIdx.x;          // 0..63
  const int b  = bh >> 3;
  const int h  = bh & 7;
  const int tid = threadIdx.x;        // 0..255

  __shared__ float sm[256];

  float v[4];
  float s = 0.0f;
#pragma unroll
  for (int k = 0; k < 4; ++k) {
    const int l = tid * 4 + k;
    float x = hurst[(b * L_DIM + l) * H_DIM + h];
    x = fminf(fmaxf(x, 0.01f), 1.0f);
    s += expf(GAMMA * x);
    v[k] = s;                         // inclusive within this thread's chunk
  }
  sm[tid] = s;
  __syncthreads();
  // inclusive scan of chunk sums
  for (int off = 1; off < 256; off <<= 1) {
    float t = (tid >= off) ? sm[tid - off] : 0.0f;
    __syncthreads();
    sm[tid] += t;
    __syncthreads();
  }
  const float excl = sm[tid] - s;     // exclusive prefix of this chunk
#pragma unroll
  for (int k = 0; k < 4; ++k)
    psi[bh * L_DIM + tid * 4 + k] = excl + v[k];
}

// ---------------------------------------------------------------------------
// Kernel 2: prior[bh][i][j] = exp(-(psi_i - psi_j)^2/2) / (rowsum + 1e-6)
// Block = 8 waves = 256 threads; owns one (bh, 16-row strip).
// Wave w owns columns [w*128, w*128+128) as 8 WMMA 16x16 tiles.
// Exponent per tile via V_WMMA_F32_16X16X4_F32:
//   A = [u, 0, 0, 0]  (K-slots 1..3 zeroed -> only B row K0 matters)
//   B row0 = v
//   C = -(u^2 + v^2)/2   (folded into accumulator, documented C layout)
//   D[m][n] = u_m * v_n - (u_m^2 + v_n^2)/2 = -(psi_i - psi_j)^2 / 2
// Tiles kept in registers until block-wide row sums are known; single store.
// ---------------------------------------------------------------------------
__global__ __launch_bounds__(256) void prior_kernel(
    const float* __restrict__ psi, float* __restrict__ out) {
  const int rowBlk = blockIdx.x;      // 0..63  (16-row strip)
  const int bh     = blockIdx.y;      // 0..63
  const int tid    = threadIdx.x;
  const int lane   = tid & 31;
  const int wave   = tid >> 5;        // 0..7
  const int half   = lane >> 4;       // 0: lanes 0-15, 1: lanes 16-31
  const int nloc   = lane & 15;

  __shared__ float s_psi[L_DIM];          // whole psi vector for this (b,h)
  __shared__ float s_part[8][16];         // per-wave partial row sums
  __shared__ float s_rowsum[16];

  const float* p = psi + (size_t)bh * L_DIM;
#pragma unroll
  for (int k = 0; k < 4; ++k) s_psi[tid + 256 * k] = p[tid + 256 * k];
  __syncthreads();

  const int i0 = rowBlk * 16;
  const float psi_i_mid = s_psi[i0 + 8];

  // psi values of the 16 strip rows, per-lane view: row m = r + half*8
  float urow[8], rowacc[8];
#pragma unroll
  for (int r = 0; r < 8; ++r) {
    urow[r]   = s_psi[i0 + r + half * 8];
    rowacc[r] = 0.0f;
  }
  const float ua_raw = s_psi[i0 + nloc];  // A-matrix row value (lanes 0-15)

  v8f e[8];                                // 8 tiles of exp values (64 VGPRs)
  const int j0base = wave * 128;
#pragma unroll
  for (int t = 0; t < 8; ++t) {
    const int j0 = j0base + t * 16;
    const float c  = 0.5f * (psi_i_mid + s_psi[j0 + 8]);  // per-tile center
    const float vj = s_psi[j0 + nloc] - c;

    v2f a, bm;
    a.x  = (lane < 16) ? (ua_raw - c) : 0.0f;  // K0 = u_m ; K2 (lanes>=16) = 0
    a.y  = 0.0f;                               // K1, K3 = 0
    bm.x = vj;                                 // B row K0 = v_n
    bm.y = 0.0f;

    v8f cm;
#pragma unroll
    for (int r = 0; r < 8; ++r) {
      const float um = urow[r] - c;
      cm[r] = -0.5f * (um * um + vj * vj);
    }

    // D = A x B + C  ->  -(psi_i - psi_j)^2 / 2
    v8f d = __builtin_amdgcn_wmma_f32_16x16x4_f32(
        false, a, false, bm, (short)0, cm, false, false);

#pragma unroll
    for (int r = 0; r < 8; ++r) {
      const float ev = __expf(d[r]);
      e[t][r]    = ev;
      rowacc[r] += ev;
    }
  }

  // reduce row partials across the 16 lanes of each half-wave
#pragma unroll
  for (int r = 0; r < 8; ++r) {
    float s = rowacc[r];
    s += __shfl_xor(s, 1, 32);
    s += __shfl_xor(s, 2, 32);
    s += __shfl_xor(s, 4, 32);
    s += __shfl_xor(s, 8, 32);
    rowacc[r] = s;
  }
  if (nloc == 0) {                         // lanes 0 and 16
#pragma unroll
    for (int r = 0; r < 8; ++r) s_part[wave][r + half * 8] = rowacc[r];
  }
  __syncthreads();
  if (tid < 16) {                          // deterministic cross-wave reduce
    float s = 0.0f;
#pragma unroll
    for (int w = 0; w < 8; ++w) s += s_part[w][tid];
    s_rowsum[tid] = s;
  }
  __syncthreads();

  float scale[8];
#pragma unroll
  for (int r = 0; r < 8; ++r)
    scale[r] = 1.0f / (s_rowsum[r + half * 8] + 1e-6f);

  // store: lane writes (row m = r + half*8, col j0 + nloc)
  float* ob = out + ((size_t)bh * L_DIM + i0) * L_DIM;
#pragma unroll
  for (int t = 0; t < 8; ++t) {
    const int j0 = j0base + t * 16;
#pragma unroll
    for (int r = 0; r < 8; ++r) {
      ob[(size_t)(r + half * 8) * L_DIM + j0 + nloc] = e[t][r] * scale[r];
    }
  }
}

extern "C" void kernel_launch(void* const* d_in, const int* in_sizes, int n_in,
                              void* d_out, int out_size, void* d_ws, size_t ws_size,
                              hipStream_t stream) {
  const float* hurst = (const float*)d_in[0];
  float* out = (float*)d_out;
  float* psi = (float*)d_ws;               // 64 * 1024 floats = 256 KB

  psi_cumsum_kernel<<<B_DIM * H_DIM, 256, 0, stream>>>(hurst, psi);

  dim3 grid(L_DIM / 16, B_DIM * H_DIM);    // (64 row strips, 64 bh)
  prior_kernel<<<grid, 256, 0, stream>>>(psi, out);
}